// VmapConvAttention_62053687492657
// MI455X (gfx1250) — compile-verified
//
#include <hip/hip_runtime.h>

typedef __attribute__((ext_vector_type(2))) float v2f;
typedef __attribute__((ext_vector_type(8))) float v8f;
typedef __attribute__((ext_vector_type(4))) float f4v;

#define F_TOTAL  128
#define N_HEADS  8
#define HEAD_DIM 16

// ---------------------------------------------------------------------------
// Kernel 1: per-head projections  q,k,v[n, h*16+e] = sum_d x[n, h*16+d] * W[h][e][d]
// One wave (32 lanes) per (16-node tile, head); 8 heads = 8 waves per 256-thread block.
// Uses V_WMMA_F32_16X16X4_F32:  D(16x16 f32) = A(16x4 f32) x B(4x16 f32) + C
// x is read exactly once -> non-temporal loads (keep L2 free for q/k/v tables).
// ---------------------------------------------------------------------------
__global__ __launch_bounds__(256) void proj_qkv_wmma(
    const float* __restrict__ x,
    const float* __restrict__ wq, const float* __restrict__ wk, const float* __restrict__ wv,
    float* __restrict__ q, float* __restrict__ k, float* __restrict__ v,
    int n_nodes)
{
    const int tile = blockIdx.x;            // 16-node tile index
    const int head = threadIdx.x >> 5;      // wave id == head (wave32)
    const int lane = threadIdx.x & 31;
    const int half = lane >> 4;             // 0: lanes 0-15, 1: lanes 16-31
    const int l15  = lane & 15;

    // --- A fragments: x tile rows. ISA 16x4 f32 A layout:
    //   lanes 0-15 : M=lane,    VGPR0=K+0, VGPR1=K+1
    //   lanes 16-31: M=lane-16, VGPR0=K+2, VGPR1=K+3
    int row = tile * 16 + l15;
    row = (row < n_nodes) ? row : (n_nodes - 1);   // clamp, no branch (EXEC stays all-1)
    const float* xrow = x + (size_t)row * F_TOTAL + head * HEAD_DIM + half * 2;
    v2f a[4];
#pragma unroll
    for (int s = 0; s < 4; ++s) {
        a[s].x = __builtin_nontemporal_load(xrow + 4 * s + 0);
        a[s].y = __builtin_nontemporal_load(xrow + 4 * s + 1);
    }

    // --- B fragments: B[kk][n] = W[n][kk]  (because y = x @ W^T)
    //   mirrored layout: lanes 0-15: N=lane, rows K+0/K+1 ; lanes 16-31: rows K+2/K+3
    const float* wqh = wq + head * 256 + l15 * 16 + half * 2;
    const float* wkh = wk + head * 256 + l15 * 16 + half * 2;
    const float* wvh = wv + head * 256 + l15 * 16 + half * 2;

    v8f cq = {0.f,0.f,0.f,0.f,0.f,0.f,0.f,0.f};
    v8f ck = cq;
    v8f cv = cq;
#pragma unroll
    for (int s = 0; s < 4; ++s) {
        v2f bq, bk, bv;
        bq.x = wqh[4 * s]; bq.y = wqh[4 * s + 1];
        bk.x = wkh[4 * s]; bk.y = wkh[4 * s + 1];
        bv.x = wvh[4 * s]; bv.y = wvh[4 * s + 1];
        cq = __builtin_amdgcn_wmma_f32_16x16x4_f32(false, a[s], false, bq, (short)0, cq, false, false);
        ck = __builtin_amdgcn_wmma_f32_16x16x4_f32(false, a[s], false, bk, (short)0, ck, false, false);
        cv = __builtin_amdgcn_wmma_f32_16x16x4_f32(false, a[s], false, bv, (short)0, cv, false, false);
    }

    // --- D layout: VGPR r -> (M = r + 8*half, N = l15)
    const int col = head * HEAD_DIM + l15;
#pragma unroll
    for (int r = 0; r < 8; ++r) {
        const int m = tile * 16 + r + 8 * half;
        if (m < n_nodes) {
            const size_t off = (size_t)m * F_TOTAL + col;
            q[off] = cq[r];
            k[off] = ck[r];
            v[off] = cv[r];
        }
    }
}

// ---------------------------------------------------------------------------
// Kernel 2: zero the scatter-add target
// ---------------------------------------------------------------------------
__global__ __launch_bounds__(256) void zero_out_kernel(float* __restrict__ out, int n_elems)
{
    int gid = blockIdx.x * blockDim.x + threadIdx.x;
    int stride = gridDim.x * blockDim.x;
    for (int i = gid; i < n_elems; i += stride)
        out[i] = 0.0f;
}

// ---------------------------------------------------------------------------
// Kernel 3: pair attention + scatter-add.  One thread per (pair, head).
//   alpha = <q[i,h,:], w[p,h,:] * k[j,h,:]> / sqrt(16) * phi[p]
//   out[i,h,:] += alpha * v[j,h,:]          (hardware global_atomic_add_f32)
// w_ij / idx / phi are streamed once -> non-temporal loads so they don't evict
// the q/k/v gather tables (~154 MB) from the 192 MB L2.
// ---------------------------------------------------------------------------
__global__ __launch_bounds__(256) void pair_attn_scatter(
    const float* __restrict__ q, const float* __restrict__ k, const float* __restrict__ v,
    const float* __restrict__ w_ij,
    const int* __restrict__ idx_i, const int* __restrict__ idx_j,
    const float* __restrict__ phi_r_cut,
    float* __restrict__ out, int n_pairs)
{
    const int gid = blockIdx.x * blockDim.x + threadIdx.x;
    const int p = gid >> 3;       // pair
    const int h = gid & 7;        // head
    if (p >= n_pairs) return;

    const int   i  = __builtin_nontemporal_load(idx_i + p);
    const int   j  = __builtin_nontemporal_load(idx_j + p);
    const float ph = __builtin_nontemporal_load(phi_r_cut + p);

    const f4v* qv = (const f4v*)(q    + (size_t)i * F_TOTAL + h * HEAD_DIM);
    const f4v* kv = (const f4v*)(k    + (size_t)j * F_TOTAL + h * HEAD_DIM);
    const f4v* wv = (const f4v*)(w_ij + (size_t)p * F_TOTAL + h * HEAD_DIM);

    float alpha = 0.0f;
#pragma unroll
    for (int t = 0; t < 4; ++t) {
        f4v a = qv[t];
        f4v b = __builtin_nontemporal_load(wv + t);   // streaming, read-once
        f4v c = kv[t];
        alpha += a.x * b.x * c.x + a.y * b.y * c.y + a.z * b.z * c.z + a.w * b.w * c.w;
    }
    alpha *= 0.25f * ph;          // 1/sqrt(d), d = 16

    const f4v* vv = (const f4v*)(v + (size_t)j * F_TOTAL + h * HEAD_DIM);
    float* op = out + (size_t)i * F_TOTAL + h * HEAD_DIM;
#pragma unroll
    for (int t = 0; t < 4; ++t) {
        f4v m = vv[t];
        unsafeAtomicAdd(op + 4 * t + 0, alpha * m.x);
        unsafeAtomicAdd(op + 4 * t + 1, alpha * m.y);
        unsafeAtomicAdd(op + 4 * t + 2, alpha * m.z);
        unsafeAtomicAdd(op + 4 * t + 3, alpha * m.w);
    }
}

// ---------------------------------------------------------------------------
extern "C" void kernel_launch(void* const* d_in, const int* in_sizes, int n_in,
                              void* d_out, int out_size, void* d_ws, size_t ws_size,
                              hipStream_t stream)
{
    const float* x         = (const float*)d_in[0];
    const float* w_ij      = (const float*)d_in[1];
    const int*   idx_i     = (const int*)  d_in[2];
    const int*   idx_j     = (const int*)  d_in[3];
    const float* phi_r_cut = (const float*)d_in[4];
    const float* wq        = (const float*)d_in[5];
    const float* wk        = (const float*)d_in[6];
    const float* wv        = (const float*)d_in[7];
    float* out = (float*)d_out;

    const int n_nodes = in_sizes[0] / F_TOTAL;
    const int n_pairs = in_sizes[2];

    // workspace: q, k, v  (each n_nodes * 128 f32)
    float* qbuf = (float*)d_ws;
    float* kbuf = qbuf + (size_t)n_nodes * F_TOTAL;
    float* vbuf = kbuf + (size_t)n_nodes * F_TOTAL;

    // zero the output (scatter-add target)
    {
        int blocks = (out_size + 256 * 8 - 1) / (256 * 8);
        if (blocks < 1) blocks = 1;
        zero_out_kernel<<<blocks, 256, 0, stream>>>(out, out_size);
    }

    // q/k/v projections via f32 WMMA
    {
        int tiles = (n_nodes + 15) / 16;
        proj_qkv_wmma<<<tiles, 256, 0, stream>>>(x, wq, wk, wv, qbuf, kbuf, vbuf, n_nodes);
    }

    // pair attention + scatter
    {
        long long threads = (long long)n_pairs * N_HEADS;
        int blocks = (int)((threads + 255) / 256);
        pair_attn_scatter<<<blocks, 256, 0, stream>>>(qbuf, kbuf, vbuf, w_ij,
                                                      idx_i, idx_j, phi_r_cut,
                                                      out, n_pairs);
    }
}